// ExpressionEncoder_37864431682197
// MI455X (gfx1250) — compile-verified
//
#include <hip/hip_runtime.h>
#include <hip/hip_bf16.h>

// ---------------------------------------------------------------------------
// ExpressionEncoder on gfx1250 (MI455X):
//   k1: per-block packed scan of (is_id, is_kos)
//   k2: exclusive scan of block totals
//   k3: resolve masked_scatter gather codes per token
//   k4: fused 2-layer MLP using V_WMMA_F32_16X16X4_F32, TM=48 (3 A-stripes
//       per wave share each B fragment -> 24 FLOP per L2 weight byte).
//       K extents are runtime args so the k-loops stay ROLLED (unroll 8)
//       -> <256 VGPRs, deep pipelined B loads. ~147 KB LDS, 2 WGs/WGP.
// ---------------------------------------------------------------------------

typedef float v2f __attribute__((ext_vector_type(2)));
typedef float v8f __attribute__((ext_vector_type(8)));

#define NTOK        65536     // B*S
#define DENC        256
#define KIND_D      8
#define XDIM        264       // KIND_D + DENC
#define EXPR_D      512
#define TM          48        // tokens per workgroup (three 16-row stripes)
#define NBLK        ((NTOK + TM - 1) / TM)   // 1366
#define XS_STRIDE   268       // 264 padded (bank-conflict-free A loads)
#define HS_STRIDE   516       // 512 padded

// ---------------- kernel 1: local packed inclusive scan --------------------
__global__ __launch_bounds__(256) void scan_local_k(
    const int* __restrict__ token_type,
    int* __restrict__ packed,
    int* __restrict__ blkId,
    int* __restrict__ blkKos)
{
    __shared__ int s[256];
    const int t = threadIdx.x;
    const int i = blockIdx.x * 256 + t;
    const int kind = token_type[i];
    const int isId  = (kind == 3) ? 1 : 0;
    const int isKos = (kind == 4 || kind == 5 || kind == 6) ? 1 : 0;
    s[t] = isId | (isKos << 16);
    __syncthreads();
    #pragma unroll
    for (int off = 1; off < 256; off <<= 1) {
        int v = (t >= off) ? s[t - off] : 0;
        __syncthreads();
        s[t] += v;
        __syncthreads();
    }
    packed[i] = s[t];
    if (t == 255) {
        blkId[blockIdx.x]  = s[255] & 0xFFFF;
        blkKos[blockIdx.x] = (s[255] >> 16) & 0xFFFF;
    }
}

// ---------------- kernel 2: exclusive scan of 256 block totals -------------
__global__ __launch_bounds__(256) void scan_blocks_k(
    const int* __restrict__ blkId,
    const int* __restrict__ blkKos,
    int* __restrict__ offId,
    int* __restrict__ offKos)
{
    __shared__ int sa[256];
    __shared__ int sb[256];
    const int t = threadIdx.x;
    sa[t] = blkId[t];
    sb[t] = blkKos[t];
    __syncthreads();
    #pragma unroll
    for (int off = 1; off < 256; off <<= 1) {
        int va = (t >= off) ? sa[t - off] : 0;
        int vb = (t >= off) ? sb[t - off] : 0;
        __syncthreads();
        sa[t] += va;
        sb[t] += vb;
        __syncthreads();
    }
    offId[t]  = (t == 0) ? 0 : sa[t - 1];
    offKos[t] = (t == 0) ? 0 : sb[t - 1];
}

// ---------------- kernel 3: resolve gather codes ---------------------------
__global__ __launch_bounds__(256) void resolve_k(
    const int* __restrict__ token_type,
    const int* __restrict__ packed,
    const int* __restrict__ offId,
    const int* __restrict__ offKos,
    const int* __restrict__ idIdx,    // identifier_indices
    const int* __restrict__ kosIdx,   // kos_token_index
    int* __restrict__ codes)
{
    const int b = blockIdx.x;
    const int i = b * 256 + threadIdx.x;
    const int kind = token_type[i];
    const int p = packed[i];
    int code = -1;
    if (kind == 3) {
        int r = offId[b] + (p & 0xFFFF) - 1;
        code = idIdx[r];                      // row in encoded_identifiers
    } else if (kind == 4 || kind == 5 || kind == 6) {
        int r = offKos[b] + ((p >> 16) & 0xFFFF) - 1;
        code = 0x40000000 | kosIdx[r];        // row in kos_emb
    }
    codes[i] = code;
}

// ---------------- kernel 4: fused MLP with FP32 WMMA -----------------------
__global__ __launch_bounds__(256) void mlp_wmma_k(
    const int*   __restrict__ token_type,
    const int*   __restrict__ codes,
    const float* __restrict__ enc_ids,   // [8192, 256]
    const float* __restrict__ kind_emb,  // [8, 8]
    const float* __restrict__ kos_emb,   // [256, 256]
    const float* __restrict__ Wp,        // [264, 512]
    const float* __restrict__ bp,        // [512]
    const float* __restrict__ Wa,        // [512, 512]
    const float* __restrict__ ba,        // [512]
    float*       __restrict__ out,       // [NTOK, 512]
    int kd1, int kd2)                    // runtime K extents (264, 512)
{
    __shared__ float xs[TM * XS_STRIDE];   // x tile  [48 x 264(+pad)]  ~50 KB
    __shared__ float hs[TM * HS_STRIDE];   // h tile  [48 x 512(+pad)]  ~97 KB

    const int tid     = threadIdx.x;
    const int tokBase = blockIdx.x * TM;

    // ---- stage x tile: kind embedding (cols 0..7) ----
    for (int u = tid; u < TM * KIND_D; u += 256) {
        const int r = u >> 3;
        const int c = u & 7;
        const int tok = tokBase + r;
        float v = 0.f;
        if (tok < NTOK) {
            const int kind = token_type[tok];
            v = kind_emb[kind * KIND_D + c];
        }
        xs[r * XS_STRIDE + c] = v;
    }
    // ---- stage x tile: enc gather (cols 8..263), float4 granularity ----
    for (int u = tid; u < TM * (DENC / 4); u += 256) {
        const int r  = u >> 6;            // token row 0..47
        const int c4 = (u & 63) << 2;     // col within enc, multiple of 4
        const int tok = tokBase + r;
        const int code = (tok < NTOK) ? codes[tok] : -1;
        float4 v = make_float4(0.f, 0.f, 0.f, 0.f);
        if (code >= 0) {
            const float* src = (code & 0x40000000)
                ? (kos_emb + (size_t)(code & 0xFFFF) * DENC)
                : (enc_ids + (size_t)code * DENC);
            v = *(const float4*)(src + c4);
        }
        *(float4*)(&xs[r * XS_STRIDE + KIND_D + c4]) = v;
    }
    __syncthreads();

    const int lane  = tid & 31;
    const int wave  = tid >> 5;
    const int m     = lane & 15;           // row within stripe / col-in-tile
    const int kh    = (lane >> 4) << 1;    // K half offset: 0 or 2
    const int rbase = (lane >> 4) << 3;    // C/D row base: 0 or 8

    // ---------------- layer 1: h = relu(x @ Wp + bp) ----------------
    for (int nt = wave; nt < EXPR_D / 16; nt += 8) {
        const int ncol = nt * 16 + m;
        const float bias = bp[ncol];
        v8f acc0, acc1, acc2;
        #pragma unroll
        for (int j = 0; j < 8; ++j) { acc0[j] = bias; acc1[j] = bias; acc2[j] = bias; }

        const float* wcol  = Wp + ncol + (size_t)kh * EXPR_D;
        const float* xrow0 = xs + m * XS_STRIDE + kh;
        const float* xrow1 = xrow0 + 16 * XS_STRIDE;
        const float* xrow2 = xrow0 + 32 * XS_STRIDE;
        #pragma clang loop unroll_count(8)
        for (int k = 0; k < kd1; k += 4) {
            v2f b;
            b.x = wcol[(size_t)k * EXPR_D];
            b.y = wcol[(size_t)k * EXPR_D + EXPR_D];
            v2f a0, a1, a2;
            a0.x = xrow0[k];  a0.y = xrow0[k + 1];
            a1.x = xrow1[k];  a1.y = xrow1[k + 1];
            a2.x = xrow2[k];  a2.y = xrow2[k + 1];
            acc0 = __builtin_amdgcn_wmma_f32_16x16x4_f32(
                false, a0, false, b, (short)0, acc0, false, false);
            acc1 = __builtin_amdgcn_wmma_f32_16x16x4_f32(
                false, a1, false, b, (short)0, acc1, false, false);
            acc2 = __builtin_amdgcn_wmma_f32_16x16x4_f32(
                false, a2, false, b, (short)0, acc2, false, false);
        }
        #pragma unroll
        for (int j = 0; j < 8; ++j) {
            float v0 = acc0[j] > 0.f ? acc0[j] : 0.f;
            float v1 = acc1[j] > 0.f ? acc1[j] : 0.f;
            float v2 = acc2[j] > 0.f ? acc2[j] : 0.f;
            hs[(rbase + j) * HS_STRIDE + ncol]      = v0;
            hs[(16 + rbase + j) * HS_STRIDE + ncol] = v1;
            hs[(32 + rbase + j) * HS_STRIDE + ncol] = v2;
        }
    }
    __syncthreads();

    // ---------------- layer 2: out = relu(h @ Wa + ba) ----------------
    for (int nt = wave; nt < EXPR_D / 16; nt += 8) {
        const int ncol = nt * 16 + m;
        const float bias = ba[ncol];
        v8f acc0, acc1, acc2;
        #pragma unroll
        for (int j = 0; j < 8; ++j) { acc0[j] = bias; acc1[j] = bias; acc2[j] = bias; }

        const float* wcol  = Wa + ncol + (size_t)kh * EXPR_D;
        const float* hrow0 = hs + m * HS_STRIDE + kh;
        const float* hrow1 = hrow0 + 16 * HS_STRIDE;
        const float* hrow2 = hrow0 + 32 * HS_STRIDE;
        #pragma clang loop unroll_count(8)
        for (int k = 0; k < kd2; k += 4) {
            v2f b;
            b.x = wcol[(size_t)k * EXPR_D];
            b.y = wcol[(size_t)k * EXPR_D + EXPR_D];
            v2f a0, a1, a2;
            a0.x = hrow0[k];  a0.y = hrow0[k + 1];
            a1.x = hrow1[k];  a1.y = hrow1[k + 1];
            a2.x = hrow2[k];  a2.y = hrow2[k + 1];
            acc0 = __builtin_amdgcn_wmma_f32_16x16x4_f32(
                false, a0, false, b, (short)0, acc0, false, false);
            acc1 = __builtin_amdgcn_wmma_f32_16x16x4_f32(
                false, a1, false, b, (short)0, acc1, false, false);
            acc2 = __builtin_amdgcn_wmma_f32_16x16x4_f32(
                false, a2, false, b, (short)0, acc2, false, false);
        }
        #pragma unroll
        for (int j = 0; j < 8; ++j) {
            const int r0 = tokBase + rbase + j;
            const int r1 = r0 + 16;
            const int r2 = r0 + 32;
            float v0 = acc0[j] > 0.f ? acc0[j] : 0.f;
            float v1 = acc1[j] > 0.f ? acc1[j] : 0.f;
            float v2 = acc2[j] > 0.f ? acc2[j] : 0.f;
            if (r0 < NTOK) out[(size_t)r0 * EXPR_D + ncol] = v0;
            if (r1 < NTOK) out[(size_t)r1 * EXPR_D + ncol] = v1;
            if (r2 < NTOK) out[(size_t)r2 * EXPR_D + ncol] = v2;
        }
    }
}

// ---------------------------------------------------------------------------
extern "C" void kernel_launch(void* const* d_in, const int* in_sizes, int n_in,
                              void* d_out, int out_size, void* d_ws, size_t ws_size,
                              hipStream_t stream) {
    const int*   token_type = (const int*)  d_in[0];
    const int*   kosIdx     = (const int*)  d_in[1];
    const int*   idIdx      = (const int*)  d_in[2];
    const float* enc_ids    = (const float*)d_in[3];
    const float* kind_emb   = (const float*)d_in[4];
    const float* kos_emb    = (const float*)d_in[5];
    const float* Wp         = (const float*)d_in[6];
    const float* bp         = (const float*)d_in[7];
    const float* Wa         = (const float*)d_in[8];
    const float* ba         = (const float*)d_in[9];
    float* out = (float*)d_out;

    int* ws      = (int*)d_ws;
    int* packed  = ws;               // 65536 ints
    int* blkId   = packed + NTOK;    // 256
    int* blkKos  = blkId + 256;      // 256
    int* offId   = blkKos + 256;     // 256
    int* offKos  = offId + 256;      // 256
    int* codes   = offKos + 256;     // 65536 ints

    scan_local_k<<<NTOK / 256, 256, 0, stream>>>(token_type, packed, blkId, blkKos);
    scan_blocks_k<<<1, 256, 0, stream>>>(blkId, blkKos, offId, offKos);
    resolve_k<<<NTOK / 256, 256, 0, stream>>>(token_type, packed, offId, offKos,
                                              idIdx, kosIdx, codes);
    mlp_wmma_k<<<NBLK, 256, 0, stream>>>(token_type, codes, enc_ids, kind_emb,
                                         kos_emb, Wp, bp, Wa, ba, out,
                                         XDIM, EXPR_D);
}